// PropertyAwareReadout_24266565222499
// MI455X (gfx1250) — compile-verified
//
#include <hip/hip_runtime.h>
#include <math.h>

typedef __attribute__((ext_vector_type(2))) float v2f;
typedef __attribute__((ext_vector_type(8))) float v8f;

#define NUM_NODES   262144
#define NUM_VARS    65536
#define HIDDEN      256
#define NUM_PROPS   8
#define NUM_GRAPHS  512

#define ROWS    64      // rows of X per block
#define KPANEL  128     // K panel staged in LDS
#define XS_LD   132     // padded LDS row stride (floats): 132 % 64 = 4 -> conflict-free A reads
#define MAXSEG  8       // LDS segment bins per block (fallback to global atomics if exceeded)

// ---- monotone float <-> uint key for atomicMax over signed floats ----
__device__ __forceinline__ unsigned fkey(float f) {
  unsigned u = __float_as_uint(f);
  return (u & 0x80000000u) ? ~u : (u | 0x80000000u);
}
__device__ __forceinline__ float fdecode(unsigned k) {
  if (k == 0u) return 0.0f;  // empty segment -> 0 (matches reference fill)
  return (k & 0x80000000u) ? __uint_as_float(k & 0x7FFFFFFFu)
                           : __uint_as_float(~k);
}

// -------------------- K0: zero accumulators --------------------
__global__ void pa_init_kernel(unsigned* __restrict__ p, int n) {
  int i = blockIdx.x * blockDim.x + threadIdx.x;
  if (i < n) p[i] = 0u;
}

// -------------------- K1: gate MLP --------------------
// gate[v] = sigmoid( relu(props[v,:] @ W1 + b1) @ W2 + b2 )
__global__ void __launch_bounds__(256)
pa_gate_kernel(const float* __restrict__ props, const float* __restrict__ W1,
               const float* __restrict__ b1, const float* __restrict__ W2,
               const float* __restrict__ b2, float* __restrict__ gate) {
  __shared__ float sW1[NUM_PROPS * 64];
  __shared__ float sb1[64];
  __shared__ float sW2[64];
  __shared__ float sb2;
  const int t = threadIdx.x;
  for (int i = t; i < NUM_PROPS * 64; i += 256) sW1[i] = W1[i];
  if (t < 64) { sb1[t] = b1[t]; sW2[t] = W2[t]; }
  if (t == 0) sb2 = b2[0];
  __syncthreads();

  const int v = blockIdx.x * 256 + t;
  float p[NUM_PROPS];
#pragma unroll
  for (int i = 0; i < NUM_PROPS; ++i) p[i] = props[(size_t)v * NUM_PROPS + i];
  float o = sb2;
#pragma unroll 4
  for (int q = 0; q < 64; ++q) {
    float a = sb1[q];
#pragma unroll
    for (int i = 0; i < NUM_PROPS; ++i) a = fmaf(p[i], sW1[i * 64 + q], a);
    o = fmaf(fmaxf(a, 0.0f), sW2[q], o);
  }
  gate[v] = 1.0f / (1.0f + expf(-o));
}

// -------------------- K2: fused GEMM (WMMA f32) + gate + segment sum/max ---
__global__ void __launch_bounds__(256)
pa_fused_kernel(const float* __restrict__ X, const int* __restrict__ batch,
                const float* __restrict__ gate, const int* __restrict__ ntype,
                const float* __restrict__ Wp, const float* __restrict__ bp,
                float* __restrict__ sums, unsigned* __restrict__ maxk,
                int* __restrict__ counts) {
  __shared__ float    Xs[ROWS * XS_LD];          // 33,792 B
  __shared__ float    sumB[MAXSEG * HIDDEN];     //  8,192 B
  __shared__ unsigned maxB[MAXSEG * HIDDEN];     //  8,192 B
  __shared__ float    wgtS[ROWS];
  __shared__ int      segS[ROWS];

  const int t = threadIdx.x;
  const int rowBase = blockIdx.x * ROWS;

  // per-row metadata: segment id, node weight, segment counts
  if (t < ROWS) {
    const int rg = rowBase + t;
    const int sg = batch[rg];
    segS[t] = sg;
    wgtS[t] = (ntype[rg] == 0) ? gate[rg >> 2] : 1.0f;  // every 4th node is a variable
    atomicAdd(&counts[sg], 1);
  }
  for (int i = t; i < MAXSEG * HIDDEN; i += 256) { sumB[i] = 0.0f; maxB[i] = 0u; }

  // wave assignment: 8 waves = 4 row-tiles x 2 col-halves
  const int ln   = t & 15;
  const int hi   = (t >> 4) & 1;   // lane >= 16 within wave32
  const int wave = t >> 5;
  const int rt   = wave & 3;       // row tile 0..3 (16 rows each)
  const int ch   = wave >> 2;      // column half 0..1 (128 cols each)
  const int mrow = rt * 16 + ln;   // A-matrix M row for this lane
  const int colB = ch * 128 + ln;  // base column for B frags

  v8f acc[8];
#pragma unroll
  for (int c = 0; c < 8; ++c)
#pragma unroll
    for (int r = 0; r < 8; ++r) acc[c][r] = 0.0f;

  const int r0 = t >> 5;
  const int c0 = (t & 31) * 4;

  for (int kp = 0; kp < HIDDEN; kp += KPANEL) {
    __syncthreads();  // previous panel fully consumed
    // stage 64x128 panel of X, coalesced float4 per thread
#pragma unroll
    for (int i = 0; i < 8; ++i) {
      const int row = r0 + 8 * i;
      const float4 v4 =
          *(const float4*)(X + (size_t)(rowBase + row) * HIDDEN + kp + c0);
      *(float4*)&Xs[row * XS_LD + c0] = v4;
    }
    __syncthreads();

    for (int kk = 0; kk < KPANEL; kk += 4) {
      // A frag (16x4 f32): VGPR v -> K = kk + v + 2*hi
      v2f a;
      a.x = Xs[mrow * XS_LD + kk + 2 * hi];
      a.y = Xs[mrow * XS_LD + kk + 2 * hi + 1];
      const float* wp = Wp + (size_t)(kp + kk + 2 * hi) * HIDDEN + colB;
#pragma unroll
      for (int c = 0; c < 8; ++c) {
        // B frag (4x16 f32): VGPR v -> K = kk + v + 2*hi, N = lane&15
        v2f b;
        b.x = wp[c * 16];
        b.y = wp[c * 16 + HIDDEN];
        acc[c] = __builtin_amdgcn_wmma_f32_16x16x4_f32(
            false, a, false, b, (short)0, acc[c], false, false);
      }
    }
  }

  // bias + gate weight + segment reduction
  const int seg0 = segS[0];
  const int nseg = segS[ROWS - 1] - seg0 + 1;
#pragma unroll
  for (int c = 0; c < 8; ++c) {
    const int col = ch * 128 + c * 16 + ln;
    const float bb = bp[col];
#pragma unroll
    for (int r = 0; r < 8; ++r) {
      const int ml = rt * 16 + hi * 8 + r;  // C/D layout: row = r + 8*(lane>=16)
      const float val = (acc[c][r] + bb) * wgtS[ml];
      if (nseg <= MAXSEG) {
        const int idx = (segS[ml] - seg0) * HIDDEN + col;
        atomicAdd(&sumB[idx], val);
        atomicMax(&maxB[idx], fkey(val));
      } else {  // pathological block spanning many segments: direct global
        const size_t gi = (size_t)segS[ml] * HIDDEN + col;
        atomicAdd(&sums[gi], val);
        atomicMax(&maxk[gi], fkey(val));
      }
    }
  }
  __syncthreads();

  if (nseg <= MAXSEG) {  // flush LDS bins to global
    for (int i = t; i < nseg * HIDDEN; i += 256) {
      const int ls = i >> 8, col = i & 255;
      const size_t gi = (size_t)(seg0 + ls) * HIDDEN + col;
      const float sv = sumB[i];
      const unsigned mk = maxB[i];
      if (sv != 0.0f) atomicAdd(&sums[gi], sv);
      if (mk != 0u) atomicMax(&maxk[gi], mk);
    }
  }
}

// -------------------- K3: finalize: [mean|max] @ W_post + b_post ----------
__global__ void __launch_bounds__(256)
pa_final_kernel(const float* __restrict__ sums, const unsigned* __restrict__ maxk,
                const int* __restrict__ counts, const float* __restrict__ Wpost,
                const float* __restrict__ bpost, float* __restrict__ out) {
  __shared__ float hc[2 * HIDDEN];
  const int s = blockIdx.x;
  const int t = threadIdx.x;
  float cnt = (float)counts[s];
  if (cnt < 1.0f) cnt = 1.0f;
  hc[t] = sums[(size_t)s * HIDDEN + t] / cnt;
  hc[HIDDEN + t] = fdecode(maxk[(size_t)s * HIDDEN + t]);
  __syncthreads();
  float a = bpost[t];
#pragma unroll 4
  for (int j = 0; j < 2 * HIDDEN; ++j)
    a = fmaf(hc[j], Wpost[(size_t)j * HIDDEN + t], a);
  out[(size_t)s * HIDDEN + t] = a;
}

// -------------------- host --------------------
extern "C" void kernel_launch(void* const* d_in, const int* in_sizes, int n_in,
                              void* d_out, int out_size, void* d_ws, size_t ws_size,
                              hipStream_t stream) {
  (void)in_sizes; (void)n_in; (void)out_size; (void)ws_size;
  const float* X     = (const float*)d_in[0];
  const int*   batch = (const int*)d_in[1];
  const float* props = (const float*)d_in[2];
  const int*   ntype = (const int*)d_in[3];
  const float* Wpre  = (const float*)d_in[4];
  const float* bpre  = (const float*)d_in[5];
  const float* W1    = (const float*)d_in[6];
  const float* b1    = (const float*)d_in[7];
  const float* W2    = (const float*)d_in[8];
  const float* b2    = (const float*)d_in[9];
  const float* Wpost = (const float*)d_in[10];
  const float* bpost = (const float*)d_in[11];
  float* out = (float*)d_out;

  // workspace layout (contiguous so one zeroing kernel covers sums..counts)
  float*    gate   = (float*)d_ws;                       // NUM_VARS
  float*    sums   = gate + NUM_VARS;                    // NUM_GRAPHS*HIDDEN
  unsigned* maxk   = (unsigned*)(sums + NUM_GRAPHS * HIDDEN);
  int*      counts = (int*)(maxk + NUM_GRAPHS * HIDDEN); // NUM_GRAPHS

  const int zn = NUM_GRAPHS * HIDDEN * 2 + NUM_GRAPHS;
  pa_init_kernel<<<(zn + 255) / 256, 256, 0, stream>>>((unsigned*)sums, zn);
  pa_gate_kernel<<<NUM_VARS / 256, 256, 0, stream>>>(props, W1, b1, W2, b2, gate);
  pa_fused_kernel<<<NUM_NODES / ROWS, 256, 0, stream>>>(
      X, batch, gate, ntype, Wpre, bpre, sums, maxk, counts);
  pa_final_kernel<<<NUM_GRAPHS, 256, 0, stream>>>(sums, maxk, counts, Wpost,
                                                  bpost, out);
}